// GCN_19662360281500
// MI455X (gfx1250) — compile-verified
//
#include <hip/hip_runtime.h>
#include <hip/hip_bf16.h>

// ---------------------------------------------------------------------------
// Problem constants (from reference): B=64, NUM_CLASS=8, H_DIM=49, G_DIM=1024
// Math collapses to:
//   mu[b,f]  = mean_n x[b,n,f]            (n over 392 nodes)       -> 103 MB read
//   m1[b,:]  = relu(mu W1g^T + b1g)       (64x1024x1024 f32 GEMM)  -> WMMA f32
//   m2[b,:]  = relu(m1 W2g^T + b2g)       (64x128x1024 f32 GEMM)   -> WMMA f32
//   s_g[b]   = mean m2[b,:]
//   m1h/m2h  = tiny 49-dim stage, s_h[b] = mean m2h[b,:]
//   m2n[b,:] = tiny 8-dim stage
//   out[b,c,h,k] = m2n[b,c]                                        -> 12.8 MB write
// Memory floor ~116MB -> ~5us @ 23.3 TB/s; FLOPs negligible.
// ---------------------------------------------------------------------------

#define BATCH   64
#define NODES_G 392          // 8*49
#define GDIM    1024
#define SPLITS  8            // split 392 rows into 8 chunks of 49 for occupancy
#define CHUNK   49

typedef __attribute__((ext_vector_type(2))) float v2f;
typedef __attribute__((ext_vector_type(8))) float v8f;

// ---- workspace layout (float offsets) ----
#define WS_PART  0                                   // 64*8*1024
#define WS_MU    (WS_PART + BATCH*SPLITS*GDIM)        // 64*1024
#define WS_M1    (WS_MU   + BATCH*GDIM)               // 64*1024
#define WS_M2    (WS_M1   + BATCH*GDIM)               // 64*128
#define WS_M2N   (WS_M2   + BATCH*128)                // 64*8

// ---------------------------------------------------------------------------
// Kernel 1: partial column sums over node axis. grid (64,8), 256 threads.
// Coalesced float4 reads: one row = 256 float4.
// ---------------------------------------------------------------------------
__global__ void colsum_part(const float* __restrict__ x, float* __restrict__ part) {
    const int b = blockIdx.x, s = blockIdx.y, t = threadIdx.x;
    const float4* __restrict__ xr =
        (const float4*)(x + (size_t)(b * NODES_G + s * CHUNK) * GDIM);
    float ax = 0.f, ay = 0.f, az = 0.f, aw = 0.f;
    #pragma unroll 7
    for (int n = 0; n < CHUNK; ++n) {
        float4 v = xr[n * (GDIM / 4) + t];
        ax += v.x; ay += v.y; az += v.z; aw += v.w;
    }
    float4 o; o.x = ax; o.y = ay; o.z = az; o.w = aw;
    ((float4*)(part + (size_t)(b * SPLITS + s) * GDIM))[t] = o;
}

// ---------------------------------------------------------------------------
// Kernel 2: fold 8 partials -> mu = sum/392. grid 64, 256 threads.
// ---------------------------------------------------------------------------
__global__ void colsum_finish(const float* __restrict__ part, float* __restrict__ mu) {
    const int b = blockIdx.x, t = threadIdx.x;
    float ax = 0.f, ay = 0.f, az = 0.f, aw = 0.f;
    #pragma unroll
    for (int s = 0; s < SPLITS; ++s) {
        float4 v = ((const float4*)(part + (size_t)(b * SPLITS + s) * GDIM))[t];
        ax += v.x; ay += v.y; az += v.z; aw += v.w;
    }
    const float inv = 1.0f / (float)NODES_G;
    float4 o; o.x = ax * inv; o.y = ay * inv; o.z = az * inv; o.w = aw * inv;
    ((float4*)(mu + (size_t)b * GDIM))[t] = o;
}

// ---------------------------------------------------------------------------
// Kernel 3/4: out[M x N] = relu(A[M x K] * W[N x K]^T + bias), M multiple of 16.
// One wave per 16x16 tile, K-loop of V_WMMA_F32_16X16X4_F32.
// A 16x4 layout (ISA 7.12.2): lanes 0-15 M=lane hold K=0(v0),K=1(v1);
//                             lanes16-31 M=lane-16 hold K=2(v0),K=3(v1).
// B 4x16: v0 lanes0-15 row K=0 (N=lane), lanes16-31 row K=2; v1: K=1 / K=3.
// C/D 16x16: VGPR r -> lanes0-15 M=r, lanes16-31 M=r+8; N = lane&15.
// ---------------------------------------------------------------------------
__global__ void gemm_relu_wmma(const float* __restrict__ A,
                               const float* __restrict__ W,
                               const float* __restrict__ bias,
                               float* __restrict__ out,
                               int K, int N, int tilesN) {
    const int wid  = (blockIdx.x * blockDim.x + threadIdx.x) >> 5;
    const int lane = threadIdx.x & 31;
    const int tm = wid / tilesN;          // batch tile
    const int tn = wid % tilesN;          // output-feature tile
    const int l15   = lane & 15;
    const int khalf = (lane >> 4) << 1;   // 0 or 2

    const float* __restrict__ arow = A + (size_t)(tm * 16 + l15) * K;  // mu row (M)
    const float* __restrict__ wrow = W + (size_t)(tn * 16 + l15) * K;  // W row (N)

    v8f c = {};
    for (int k0 = 0; k0 < K; k0 += 4) {
        v2f a, b;
        a.x = arow[k0 + khalf];     a.y = arow[k0 + khalf + 1];
        b.x = wrow[k0 + khalf];     b.y = wrow[k0 + khalf + 1];
        c = __builtin_amdgcn_wmma_f32_16x16x4_f32(
                /*neg_a=*/false, a, /*neg_b=*/false, b,
                /*c_mod=*/(short)0, c, /*reuse_a=*/false, /*reuse_b=*/false);
    }

    const int   ncol = tn * 16 + l15;
    const float bv   = bias[ncol];
    const int   mbase = tm * 16 + ((lane >> 4) << 3);
    #pragma unroll
    for (int r = 0; r < 8; ++r) {
        float v = fmaxf(c[r] + bv, 0.0f);
        out[(size_t)(mbase + r) * N + ncol] = v;
    }
}

// ---------------------------------------------------------------------------
// Kernel 5: per-batch scalar tail (stages h and num). grid 64, 64 threads.
// ---------------------------------------------------------------------------
__global__ void tail_stages(const float* __restrict__ m2,
                            const float* __restrict__ W1h, const float* __restrict__ b1h,
                            const float* __restrict__ W2h, const float* __restrict__ b2h,
                            const float* __restrict__ W1n, const float* __restrict__ b1n,
                            const float* __restrict__ W2n, const float* __restrict__ b2n,
                            float* __restrict__ m2n) {
    __shared__ float red[64];
    __shared__ float sm1h[49], sm2h[49], sm1n[8];
    __shared__ float s_scalar;
    const int b = blockIdx.x, t = threadIdx.x;

    // s_g = mean of m2[b, 0..127]
    red[t] = m2[b * 128 + t] + m2[b * 128 + 64 + t];
    __syncthreads();
    for (int o = 32; o > 0; o >>= 1) {
        if (t < o) red[t] += red[t + o];
        __syncthreads();
    }
    const float s_g = red[0] * (1.0f / 128.0f);

    // stage h, layer 1: m1h[g] = relu(s_g * rowsum(W1h)[g] + b1h[g])
    if (t < 49) {
        float rs = 0.f;
        for (int f = 0; f < 49; ++f) rs += W1h[t * 49 + f];
        sm1h[t] = fmaxf(s_g * rs + b1h[t], 0.0f);
    }
    __syncthreads();
    // stage h, layer 2
    if (t < 49) {
        float acc = b2h[t];
        for (int f = 0; f < 49; ++f) acc += sm1h[f] * W2h[t * 49 + f];
        sm2h[t] = fmaxf(acc, 0.0f);
    }
    __syncthreads();
    if (t == 0) {
        float s = 0.f;
        for (int i = 0; i < 49; ++i) s += sm2h[i];
        s_scalar = s * (1.0f / 49.0f);
    }
    __syncthreads();
    const float s_h = s_scalar;

    // stage num
    if (t < 8) {
        float rs = 0.f;
        for (int f = 0; f < 8; ++f) rs += W1n[t * 8 + f];
        sm1n[t] = fmaxf(s_h * rs + b1n[t], 0.0f);
    }
    __syncthreads();
    if (t < 8) {
        float acc = b2n[t];
        for (int f = 0; f < 8; ++f) acc += sm1n[f] * W2n[t * 8 + f];
        m2n[b * 8 + t] = fmaxf(acc, 0.0f);
    }
}

// ---------------------------------------------------------------------------
// Kernel 6: out[b,c,h,k] = m2n[b,c]; float4 stores. 3,211,264 floats.
// float4 index i covers (b,c) block of 49*128/4 = 1568 float4s.
// ---------------------------------------------------------------------------
__global__ void broadcast_out(const float* __restrict__ m2n, float* __restrict__ out) {
    const int i = blockIdx.x * blockDim.x + threadIdx.x;   // 0 .. 802815
    const float v = m2n[i / 1568];
    float4 o; o.x = v; o.y = v; o.z = v; o.w = v;
    ((float4*)out)[i] = o;
}

// ---------------------------------------------------------------------------
extern "C" void kernel_launch(void* const* d_in, const int* in_sizes, int n_in,
                              void* d_out, int out_size, void* d_ws, size_t ws_size,
                              hipStream_t stream) {
    const float* x   = (const float*)d_in[0];
    const float* W1g = (const float*)d_in[1];
    const float* b1g = (const float*)d_in[2];
    const float* W2g = (const float*)d_in[3];
    const float* b2g = (const float*)d_in[4];
    const float* W1h = (const float*)d_in[5];
    const float* b1h = (const float*)d_in[6];
    const float* W2h = (const float*)d_in[7];
    const float* b2h = (const float*)d_in[8];
    const float* W1n = (const float*)d_in[9];
    const float* b1n = (const float*)d_in[10];
    const float* W2n = (const float*)d_in[11];
    const float* b2n = (const float*)d_in[12];
    float* out = (float*)d_out;
    float* ws  = (float*)d_ws;

    float* part = ws + WS_PART;
    float* mu   = ws + WS_MU;
    float* m1   = ws + WS_M1;
    float* m2   = ws + WS_M2;
    float* m2n  = ws + WS_M2N;

    // 1) column-sum partials over node axis (dominant 103 MB read)
    colsum_part<<<dim3(BATCH, SPLITS), 256, 0, stream>>>(x, part);
    // 2) fold partials -> mu
    colsum_finish<<<BATCH, 256, 0, stream>>>(part, mu);
    // 3) m1 = relu(mu W1g^T + b1g): 4x64 = 256 tiles -> 32 blocks * 8 waves
    gemm_relu_wmma<<<32, 256, 0, stream>>>(mu, W1g, b1g, m1, GDIM, GDIM, 64);
    // 4) m2 = relu(m1 W2g^T + b2g): 4x8 = 32 tiles -> 4 blocks * 8 waves
    gemm_relu_wmma<<<4, 256, 0, stream>>>(m1, W2g, b2g, m2, GDIM, 128, 8);
    // 5) tiny stages h + num
    tail_stages<<<BATCH, 64, 0, stream>>>(m2, W1h, b1h, W2h, b2h,
                                          W1n, b1n, W2n, b2n, m2n);
    // 6) broadcast to output (12.8 MB write)
    broadcast_out<<<(BATCH * 8 * 49 * 128 / 4) / 256, 256, 0, stream>>>(m2n, out);
}